// MAEST_Encoder_60756607369795
// MI455X (gfx1250) — compile-verified
//
#include <hip/hip_runtime.h>
#include <hip/hip_bf16.h>

// ---------------------------------------------------------------------------
// MAEST encoder for MI455X (gfx1250).  Wave32 + V_WMMA_F32_16X16X4_F32 +
// GLOBAL_LOAD_ASYNC_TO_LDS (ASYNCcnt) staging with double-buffered prefetch.
//   tem:  ret[b,t,n,:] = sum_t' M_n[t,t'] x[b,t',n,:]   (M_n = A_n^T A_n, 12x12)
//   tem:  x = leaky(ret @ W_bt[b,t] + bias_bt[b,t] + x)           (WMMA GEMM)
//   spa:  adj[b,t,:,:] materialized; hyp = leaky(adj @ x)         (WMMA GEMM)
//         ret = leaky(adj^T @ hyp)                                (WMMA GEMM)
//         x = leaky(ret @ W_n[n] + bias_n[n] + x)                 (WMMA GEMM)
// LDS strides: A tiles stride 68 (4*odd mod 64 -> conflict-free A-frag reads,
// 272B rows 16B-aligned); B tiles stride 72 (2*72 mod 64 = 16 -> the two lane
// halves hit disjoint bank ranges; 288B rows 16B-aligned).
// ---------------------------------------------------------------------------

#define Bc 64
#define Tc 12
#define Nc 1024
#define Dc 64
#define EDc 16
#define Hc 32
#define SLOPEc 0.01f

typedef __attribute__((ext_vector_type(2))) float v2f;
typedef __attribute__((ext_vector_type(8))) float v8f;
typedef int v4i __attribute__((vector_size(16)));

#if defined(__HIP_DEVICE_COMPILE__) &&                                   \
    __has_builtin(__builtin_amdgcn_global_load_async_to_lds_b32) &&      \
    __has_builtin(__builtin_amdgcn_global_load_async_to_lds_b128)
#define USE_ASYNC 1
#else
#define USE_ASYNC 0
#endif

typedef __attribute__((address_space(1))) int* g32p;
typedef __attribute__((address_space(3))) int* l32p;
typedef __attribute__((address_space(1))) v4i* g128p;
typedef __attribute__((address_space(3))) v4i* l128p;

__device__ __forceinline__ void cp_async_b32(const float* g, float* l) {
#if USE_ASYNC
  __builtin_amdgcn_global_load_async_to_lds_b32(
      (g32p)(unsigned long long)g,
      (l32p)(unsigned int)(unsigned long long)l, 0, 0);
#else
  *l = *g;
#endif
}

__device__ __forceinline__ void cp_async_b128(const float* g, float* l) {
#if USE_ASYNC
  __builtin_amdgcn_global_load_async_to_lds_b128(
      (g128p)(unsigned long long)g,
      (l128p)(unsigned int)(unsigned long long)l, 0, 0);
#else
  *(float4*)l = *(const float4*)g;
#endif
}

__device__ __forceinline__ void cp_async_wait() {
#if USE_ASYNC
#if __has_builtin(__builtin_amdgcn_s_wait_asynccnt)
  __builtin_amdgcn_s_wait_asynccnt(0);
#else
  asm volatile("s_wait_asynccnt 0x0" ::: "memory");
#endif
#endif
}

__device__ __forceinline__ float leaky(float v) { return v >= 0.f ? v : SLOPEc * v; }

// D(16x16) += A(16xK) * B(Kx16), A/B staged in LDS, K multiple of 4.
// CDNA5 fp32 fragment layout:
//   A: lane%16 = row M; elem v holds K = kb + 2*(lane>=16) + v
//   B: lane%16 = col N; elem v holds K = kb + 2*(lane>=16) + v
//   C/D: elem r -> row M = r + 8*(lane>=16), col N = lane%16
__device__ __forceinline__ v8f wmma_f32_tile(const float* __restrict__ As, int lda,
                                             const float* __restrict__ Bs, int ldb,
                                             int K, v8f c) {
  const int lane = threadIdx.x & 31;
  const int row  = lane & 15;
  const int half = lane >> 4;
#pragma unroll 4
  for (int kb = 0; kb < K; kb += 4) {
    v2f a, b;
    a.x = As[row * lda + kb + 2 * half + 0];
    a.y = As[row * lda + kb + 2 * half + 1];
    b.x = Bs[(kb + 2 * half + 0) * ldb + row];
    b.y = Bs[(kb + 2 * half + 1) * ldb + row];
    c = __builtin_amdgcn_wmma_f32_16x16x4_f32(false, a, false, b, (short)0, c,
                                              false, false);
  }
  return c;
}

// ---------------------------------------------------------------------------
// Tiny time-feature MLP: one thread per (b,t) sample.
__global__ void k_timefeat(const float* __restrict__ src,
                           const float* __restrict__ w_day, const float* __restrict__ b_day,
                           const float* __restrict__ w_week, const float* __restrict__ b_week,
                           const float* __restrict__ w1, const float* __restrict__ b1,
                           const float* __restrict__ w2, const float* __restrict__ b2,
                           const float* __restrict__ w, const float* __restrict__ b,
                           float* __restrict__ out) {
  int s = blockIdx.x * blockDim.x + threadIdx.x;
  if (s >= Bc * Tc) return;
  float t0 = src[(size_t)s * Nc * 3 + 1];  // source[b,t,0,1]
  float t1 = src[(size_t)s * Nc * 3 + 2];  // source[b,t,0,2]
  float h0[EDc], h1[EDc];
#pragma unroll
  for (int j = 0; j < EDc; j++)
    h0[j] = t0 * w_day[j] + b_day[j] + t1 * w_week[j] + b_week[j];
#pragma unroll
  for (int j = 0; j < EDc; j++) {
    float a = b1[j];
    for (int k = 0; k < EDc; k++) a += h0[k] * w1[k * EDc + j];
    h1[j] = fmaxf(a, 0.f);
  }
#pragma unroll
  for (int j = 0; j < EDc; j++) {
    float a = b2[j];
    for (int k = 0; k < EDc; k++) a += h1[k] * w2[k * EDc + j];
    h0[j] = fmaxf(a, 0.f);
  }
#pragma unroll
  for (int j = 0; j < EDc; j++) {
    float a = b[j];
    for (int k = 0; k < EDc; k++) a += h0[k] * w[k * EDc + j];
    out[s * EDc + j] = a;
  }
}

// ---------------------------------------------------------------------------
// Per-node Gram matrix M_n = A_n^T A_n, A_n[h,t] = sum_k emb[n,k]*padj[k,h,t].
__global__ void k_mtem(const float* __restrict__ emb, const float* __restrict__ padj,
                       float* __restrict__ Mout) {
  int n = blockIdx.x;
  __shared__ float A[Hc * Tc];
  __shared__ float e[EDc];
  int tid = threadIdx.x;
  if (tid < EDc) e[tid] = emb[n * EDc + tid];
  __syncthreads();
  for (int i = tid; i < Hc * Tc; i += blockDim.x) {
    int h = i / Tc, t = i % Tc;
    float a = 0.f;
#pragma unroll
    for (int k = 0; k < EDc; k++) a += e[k] * padj[(k * Hc + h) * Tc + t];
    A[i] = a;
  }
  __syncthreads();
  for (int i = tid; i < Tc * Tc; i += blockDim.x) {
    int t = i / Tc, tp = i % Tc;
    float m = 0.f;
#pragma unroll
    for (int h = 0; h < Hc; h++) m += A[h * Tc + t] * A[h * Tc + tp];
    Mout[n * Tc * Tc + i] = m;
  }
}

// ret[b,t,n,d] = sum_t' M_n[t,t'] * x[b,t',n,d].  Block = (b,n), thread = d.
__global__ void k_ret_tem(const float* __restrict__ x, const float* __restrict__ Mn,
                          float* __restrict__ ret) {
  int b = blockIdx.x / Nc, n = blockIdx.x % Nc;
  int d = threadIdx.x;
  __shared__ float M[Tc * Tc];
  for (int i = threadIdx.x; i < Tc * Tc; i += blockDim.x) M[i] = Mn[n * Tc * Tc + i];
  __syncthreads();
  float xv[Tc];
  size_t base = ((size_t)b * Tc * Nc + n) * Dc + d;
#pragma unroll
  for (int t = 0; t < Tc; t++) xv[t] = x[base + (size_t)t * Nc * Dc];
#pragma unroll
  for (int t = 0; t < Tc; t++) {
    float a = 0.f;
#pragma unroll
    for (int tp = 0; tp < Tc; tp++) a += M[t * Tc + tp] * xv[tp];
    ret[base + (size_t)t * Nc * Dc] = a;
  }
}

// W_bt[bt,i,o] = sum_k teb[bt,k] * pool[k,i,o].  16 blocks cover io=4096.
__global__ void k_wbt(const float* __restrict__ teb, const float* __restrict__ pool,
                      float* __restrict__ Wout) {
  int io = blockIdx.x * blockDim.x + threadIdx.x;
  float pk[EDc];
#pragma unroll
  for (int k = 0; k < EDc; k++) pk[k] = pool[k * Dc * Dc + io];
  for (int bt = 0; bt < Bc * Tc; bt++) {
    float a = 0.f;
#pragma unroll
    for (int k = 0; k < EDc; k++) a += teb[bt * EDc + k] * pk[k];
    Wout[(size_t)bt * Dc * Dc + io] = a;
  }
}

__global__ void k_bbt(const float* __restrict__ teb, const float* __restrict__ bp,
                      float* __restrict__ bout) {
  int bt = blockIdx.x, o = threadIdx.x;
  float a = 0.f;
#pragma unroll
  for (int k = 0; k < EDc; k++) a += teb[bt * EDc + k] * bp[k * Dc + o];
  bout[bt * Dc + o] = a;
}

// x[bt,n0..n0+127,:] = leaky(ret_tile @ W_bt + bias + x)   -- WMMA GEMM
__global__ void k_out_tem(const float* __restrict__ ret, const float* __restrict__ Wbt,
                          const float* __restrict__ bias, float* __restrict__ x) {
  int bt = blockIdx.y;
  int n0 = blockIdx.x * 128;
  __shared__ __align__(16) float As[128 * 68];
  __shared__ __align__(16) float Ws[64 * 72];
  __shared__ float bs[64];
  int tid = threadIdx.x;
  size_t abase = ((size_t)bt * Nc + n0) * Dc;
  for (int i = tid; i < 128 * 16; i += 256) {
    int r = i >> 4, c4 = (i & 15) << 2;
    cp_async_b128(&ret[abase + (size_t)r * Dc + c4], &As[r * 68 + c4]);
  }
  for (int i = tid; i < 64 * 16; i += 256) {
    int r = i >> 4, c4 = (i & 15) << 2;
    cp_async_b128(&Wbt[(size_t)bt * Dc * Dc + r * Dc + c4], &Ws[r * 72 + c4]);
  }
  if (tid < 64) bs[tid] = bias[bt * Dc + tid];
  cp_async_wait();
  __syncthreads();
  int wv = tid >> 5, lane = tid & 31;
  int r0 = wv * 16;
  int half = lane >> 4, col = lane & 15;
#pragma unroll
  for (int ot = 0; ot < 4; ot++) {
    v8f c = {};
    c = wmma_f32_tile(&As[r0 * 68], 68, &Ws[ot * 16], 72, 64, c);
    int o = ot * 16 + col;
#pragma unroll
    for (int r = 0; r < 8; r++) {
      int n = n0 + r0 + r + 8 * half;
      size_t idx = ((size_t)bt * Nc + n) * Dc + o;
      float v = c[r] + bs[o] + x[idx];
      x[idx] = leaky(v);
    }
  }
}

// adj[bt,h,n] = sum_k teb[bt,k] * padj[k,h,n].  128 blocks cover hn=H*N.
__global__ void k_adj_spa(const float* __restrict__ teb, const float* __restrict__ padj,
                          float* __restrict__ adj) {
  int hn = blockIdx.x * blockDim.x + threadIdx.x;
  float pk[EDc];
#pragma unroll
  for (int k = 0; k < EDc; k++) pk[k] = padj[(size_t)k * Hc * Nc + hn];
  for (int bt = 0; bt < Bc * Tc; bt++) {
    float a = 0.f;
#pragma unroll
    for (int k = 0; k < EDc; k++) a += teb[bt * EDc + k] * pk[k];
    adj[(size_t)bt * Hc * Nc + hn] = a;
  }
}

// hyp[bt,:,:] = leaky(adj_bt(32x1024) @ x_bt(1024x64))
// WMMA with double-buffered async-to-LDS prefetch over 16 K-chunks of 64.
__global__ void k_hyp_spa(const float* __restrict__ adj, const float* __restrict__ x,
                          float* __restrict__ hyp) {
  int bt = blockIdx.x;
  __shared__ __align__(16) float Aj[2][32 * 68];
  __shared__ __align__(16) float Xs[2][64 * 72];
  int tid = threadIdx.x;
  int wv = tid >> 5, lane = tid & 31;
  int htile = wv >> 2, dtile = wv & 3;
  const float* adjb = adj + (size_t)bt * Hc * Nc;
  const float* xb = x + (size_t)bt * Nc * Dc;

  auto stage = [&](int n0, int buf) {
    for (int i = tid; i < 32 * 16; i += 256) {
      int h = i >> 4, c4 = (i & 15) << 2;
      cp_async_b128(&adjb[(size_t)h * Nc + n0 + c4], &Aj[buf][h * 68 + c4]);
    }
    for (int i = tid; i < 64 * 16; i += 256) {
      int nn = i >> 4, c4 = (i & 15) << 2;
      cp_async_b128(&xb[(size_t)(n0 + nn) * Dc + c4], &Xs[buf][nn * 72 + c4]);
    }
  };

  stage(0, 0);
  cp_async_wait();
  __syncthreads();
  v8f c = {};
  for (int ch = 0; ch < Nc / 64; ch++) {
    int cur = ch & 1;
    if (ch + 1 < Nc / 64) stage((ch + 1) * 64, cur ^ 1);   // prefetch next chunk
    c = wmma_f32_tile(&Aj[cur][htile * 16 * 68], 68, &Xs[cur][dtile * 16], 72, 64, c);
    cp_async_wait();
    __syncthreads();
  }
  int half = lane >> 4, col = lane & 15;
  int d = dtile * 16 + col;
  float* hb = hyp + (size_t)bt * Hc * Dc;
#pragma unroll
  for (int r = 0; r < 8; r++) {
    int h = htile * 16 + r + 8 * half;
    hb[h * Dc + d] = leaky(c[r]);
  }
}

// ret[bt,n0..n0+127,:] = leaky(adj^T(128x32) @ hyp(32x64))  -- WMMA
__global__ void k_ret_spa(const float* __restrict__ adj, const float* __restrict__ hyp,
                          float* __restrict__ ret) {
  int bt = blockIdx.y;
  int n0 = blockIdx.x * 128;
  __shared__ float At[128 * 33];
  __shared__ __align__(16) float Hs[32 * 72];
  int tid = threadIdx.x;
  const float* adjb = adj + (size_t)bt * Hc * Nc;
  for (int i = tid; i < 128 * 32; i += 256) {  // transpose scatter into LDS
    int nl = i & 127, h = i >> 7;
    cp_async_b32(&adjb[(size_t)h * Nc + n0 + nl], &At[nl * 33 + h]);
  }
  for (int i = tid; i < 32 * 16; i += 256) {
    int h = i >> 4, c4 = (i & 15) << 2;
    cp_async_b128(&hyp[(size_t)bt * Hc * Dc + h * Dc + c4], &Hs[h * 72 + c4]);
  }
  cp_async_wait();
  __syncthreads();
  int wv = tid >> 5, lane = tid & 31;
  int r0 = wv * 16;
  int half = lane >> 4, col = lane & 15;
#pragma unroll
  for (int ot = 0; ot < 4; ot++) {
    v8f c = {};
    c = wmma_f32_tile(&At[r0 * 33], 33, &Hs[ot * 16], 72, 32, c);
    int d = ot * 16 + col;
#pragma unroll
    for (int r = 0; r < 8; r++) {
      int n = n0 + r0 + r + 8 * half;
      ret[((size_t)bt * Nc + n) * Dc + d] = leaky(c[r]);
    }
  }
}

// W_n[n,i,o] = sum_k emb[n,k] * pool[k,i,o]
__global__ void k_wn(const float* __restrict__ emb, const float* __restrict__ pool,
                     float* __restrict__ Wout) {
  int io = blockIdx.x * blockDim.x + threadIdx.x;
  float pk[EDc];
#pragma unroll
  for (int k = 0; k < EDc; k++) pk[k] = pool[k * Dc * Dc + io];
  for (int n = 0; n < Nc; n++) {
    float a = 0.f;
#pragma unroll
    for (int k = 0; k < EDc; k++) a += emb[n * EDc + k] * pk[k];
    Wout[(size_t)n * Dc * Dc + io] = a;
  }
}

__global__ void k_bn(const float* __restrict__ emb, const float* __restrict__ bp,
                     float* __restrict__ bout) {
  int n = blockIdx.x, o = threadIdx.x;
  float a = 0.f;
#pragma unroll
  for (int k = 0; k < EDc; k++) a += emb[n * EDc + k] * bp[k * Dc + o];
  bout[n * Dc + o] = a;
}

// x[bt0..bt0+127, n, :] = leaky(ret_rows @ W_n + bias_n + x)  -- WMMA per node
__global__ void k_out_spa(const float* __restrict__ ret, const float* __restrict__ Wn,
                          const float* __restrict__ bn, float* __restrict__ x) {
  int n = blockIdx.y;
  int bt0 = blockIdx.x * 128;
  __shared__ __align__(16) float As[128 * 68];
  __shared__ __align__(16) float Ws[64 * 72];
  __shared__ float bs[64];
  int tid = threadIdx.x;
  for (int i = tid; i < 64 * 16; i += 256) {
    int r = i >> 4, c4 = (i & 15) << 2;
    cp_async_b128(&Wn[(size_t)n * Dc * Dc + r * Dc + c4], &Ws[r * 72 + c4]);
  }
  for (int i = tid; i < 128 * 16; i += 256) {
    int r = i >> 4, c4 = (i & 15) << 2;
    cp_async_b128(&ret[((size_t)(bt0 + r) * Nc + n) * Dc + c4], &As[r * 68 + c4]);
  }
  if (tid < 64) bs[tid] = bn[n * Dc + tid];
  cp_async_wait();
  __syncthreads();
  int wv = tid >> 5, lane = tid & 31;
  int r0 = wv * 16;
  int half = lane >> 4, col = lane & 15;
#pragma unroll
  for (int ot = 0; ot < 4; ot++) {
    v8f c = {};
    c = wmma_f32_tile(&As[r0 * 68], 68, &Ws[ot * 16], 72, 64, c);
    int o = ot * 16 + col;
#pragma unroll
    for (int r = 0; r < 8; r++) {
      int bt = bt0 + r0 + r + 8 * half;
      size_t idx = ((size_t)bt * Nc + n) * Dc + o;
      float v = c[r] + bs[o] + x[idx];
      x[idx] = leaky(v);
    }
  }
}

// ---------------------------------------------------------------------------
extern "C" void kernel_launch(void* const* d_in, const int* in_sizes, int n_in,
                              void* d_out, int out_size, void* d_ws, size_t ws_size,
                              hipStream_t stream) {
  (void)in_sizes; (void)n_in; (void)out_size; (void)ws_size;
  const float* const* p = (const float* const*)d_in;
  // setup_inputs() insertion order:
  // 0 source, 1 x_in, 2 node_emb, 3 node_emb_spg,
  // 4..13 tf1 {w_day,b_day,w_week,b_week,w1,b1,w2,b2,w,b}, 14..23 tf1_,
  // 24..26 tem1 {adj,weights_pool,bias_pool}, 27..29 tem2, 30..32 tem3,
  // 33..35 spa1 {adj,weights_pool,bias_pool}, 36..38 spa2, 39..41 spa3
  const float* src = p[0];
  const float* x_in = p[1];
  const float* node_emb = p[2];
  const float* node_emb_spg = p[3];

  float* ws = (float*)d_ws;
  float* teb1 = ws;
  float* teb2 = teb1 + (size_t)Bc * Tc * EDc;
  float* Mtem = teb2 + (size_t)Bc * Tc * EDc;
  float* Wbt = Mtem + (size_t)Nc * Tc * Tc;
  float* bbt = Wbt + (size_t)Bc * Tc * Dc * Dc;
  float* Wn = bbt + (size_t)Bc * Tc * Dc;
  float* bn = Wn + (size_t)Nc * Dc * Dc;
  float* hyp = bn + (size_t)Nc * Dc;
  float* adj = hyp + (size_t)Bc * Tc * Hc * Dc;
  float* ret = adj + (size_t)Bc * Tc * Hc * Nc;
  float* x = (float*)d_out;

  (void)hipMemcpyAsync(x, x_in, (size_t)Bc * Tc * Nc * Dc * sizeof(float),
                       hipMemcpyDeviceToDevice, stream);

  k_timefeat<<<12, 64, 0, stream>>>(src, p[4], p[5], p[6], p[7], p[8], p[9],
                                    p[10], p[11], p[12], p[13], teb1);
  k_timefeat<<<12, 64, 0, stream>>>(src, p[14], p[15], p[16], p[17], p[18], p[19],
                                    p[20], p[21], p[22], p[23], teb2);

  const int temBase[3] = {24, 27, 30};
  const int spaBase[3] = {33, 36, 39};
  for (int l = 0; l < 3; l++) {
    // ---- temporal hypergraph layer ----
    k_mtem<<<Nc, 192, 0, stream>>>(node_emb, p[temBase[l] + 0], Mtem);
    k_ret_tem<<<Bc * Nc, 64, 0, stream>>>(x, Mtem, ret);
    k_wbt<<<16, 256, 0, stream>>>(teb1, p[temBase[l] + 1], Wbt);
    k_bbt<<<Bc * Tc, 64, 0, stream>>>(teb1, p[temBase[l] + 2], bbt);
    k_out_tem<<<dim3(Nc / 128, Bc * Tc), 256, 0, stream>>>(ret, Wbt, bbt, x);
    // ---- spatial hypergraph layer ----
    k_adj_spa<<<Hc * Nc / 256, 256, 0, stream>>>(teb2, p[spaBase[l] + 0], adj);
    k_hyp_spa<<<Bc * Tc, 256, 0, stream>>>(adj, x, hyp);
    k_ret_spa<<<dim3(Nc / 128, Bc * Tc), 256, 0, stream>>>(adj, hyp, ret);
    k_wn<<<16, 256, 0, stream>>>(node_emb_spg, p[spaBase[l] + 1], Wn);
    k_bn<<<Nc, 64, 0, stream>>>(node_emb_spg, p[spaBase[l] + 2], bn);
    k_out_spa<<<dim3(Bc * Tc / 128, Nc), 256, 0, stream>>>(ret, Wn, bn, x);
  }
}